// DDINOLossL1_40553081209171
// MI455X (gfx1250) — compile-verified
//
#include <hip/hip_runtime.h>
#include <math.h>

#define K_DIM 4096
#define FEA   384
#define BATCH 32
#define NG    196
#define NL    36
#define SROWS 21760   // B * (2*196 + 8*36)
#define TROWS 12544   // 2*B*196
#define SCLS  320
#define TCLS  64

typedef __attribute__((ext_vector_type(2))) float v2f;
typedef __attribute__((ext_vector_type(8))) float v8f;

__device__ __forceinline__ float temp_from_epoch(int e) {
    // TEMP_SCHED: linspace(0.04, 0.07, 30) then 0.07
    return (e >= 30) ? 0.07f : (0.04f + (0.03f / 29.0f) * (float)e);
}

__device__ __forceinline__ void decode_srow(int row, int& crop, int& b, int& S) {
    if (row < 2 * BATCH * NG) {
        crop = row / (BATCH * NG);
        int w = row % (BATCH * NG);
        b = w / NG; S = NG;
    } else {
        int w = row - 2 * BATCH * NG;
        crop = 2 + w / (BATCH * NL);
        int w2 = w % (BATCH * NL);
        b = w2 / NL; S = NL;
    }
}

// ---------------- per-row softmax stats: teacher (center-subtracted, /temp) ----
__global__ void k_teacher_stats(const float* __restrict__ tcls,
                                const float* __restrict__ treg,
                                const float* __restrict__ center,
                                const float* __restrict__ cgrid,
                                const int* __restrict__ epoch,
                                float* __restrict__ tm, float* __restrict__ tinvZ) {
    int r = blockIdx.x;
    const float* x; const float* c;
    if (r < TCLS) { x = tcls + (size_t)r * K_DIM; c = center; }
    else          { x = treg + (size_t)(r - TCLS) * K_DIM; c = cgrid; }
    float invT = 1.0f / temp_from_epoch(epoch[0]);
    __shared__ float sm[256];
    int t = threadIdx.x;
    float mx = -INFINITY;
    for (int k = t; k < K_DIM; k += 256) mx = fmaxf(mx, (x[k] - c[k]) * invT);
    sm[t] = mx; __syncthreads();
    for (int s = 128; s > 0; s >>= 1) { if (t < s) sm[t] = fmaxf(sm[t], sm[t + s]); __syncthreads(); }
    mx = sm[0]; __syncthreads();
    float z = 0.f;
    for (int k = t; k < K_DIM; k += 256) z += __expf((x[k] - c[k]) * invT - mx);
    sm[t] = z; __syncthreads();
    for (int s = 128; s > 0; s >>= 1) { if (t < s) sm[t] += sm[t + s]; __syncthreads(); }
    if (t == 0) { tm[r] = mx; tinvZ[r] = 1.0f / sm[0]; }
}

// ---------------- per-row log-sum-exp of student logits (x*10) -----------------
__global__ void k_student_lse(const float* __restrict__ scls,
                              const float* __restrict__ sreg,
                              float* __restrict__ slse) {
    int r = blockIdx.x;
    const float* x = (r < SCLS) ? (scls + (size_t)r * K_DIM)
                                : (sreg + (size_t)(r - SCLS) * K_DIM);
    __shared__ float sm[256];
    int t = threadIdx.x;
    float mx = -INFINITY;
    for (int k = t; k < K_DIM; k += 256) mx = fmaxf(mx, x[k] * 10.0f);
    sm[t] = mx; __syncthreads();
    for (int s = 128; s > 0; s >>= 1) { if (t < s) sm[t] = fmaxf(sm[t], sm[t + s]); __syncthreads(); }
    mx = sm[0]; __syncthreads();
    float z = 0.f;
    for (int k = t; k < K_DIM; k += 256) z += __expf(x[k] * 10.0f - mx);
    sm[t] = z; __syncthreads();
    for (int s = 128; s > 0; s >>= 1) { if (t < s) sm[t] += sm[t + s]; __syncthreads(); }
    if (t == 0) slse[r] = mx + __logf(sm[0]);
}

// ---------------- teacher feature inverse L2 norms -----------------------------
__global__ void k_tfea_norm(const float* __restrict__ tfea, float* __restrict__ tfinv) {
    int r = blockIdx.x;
    const float* x = tfea + (size_t)r * FEA;
    __shared__ float sm[128];
    int t = threadIdx.x;
    float ss = 0.f;
    for (int k = t; k < FEA; k += 128) { float v = x[k]; ss += v * v; }
    sm[t] = ss; __syncthreads();
    for (int s = 64; s > 0; s >>= 1) { if (t < s) sm[t] += sm[t + s]; __syncthreads(); }
    if (t == 0) tfinv[r] = 1.0f / fmaxf(sqrtf(sm[0]), 1e-12f);
}

// ---------------- WMMA cosine-sim argmax ---------------------------------------
// One wave per (teacher view t, batch b, 16-row student tile).
// C = A(16 x 384 student fea) * B(384 x 196 teacher fea^T), argmax over cols.
// Student row norm is a positive per-row scale -> argmax needs only teacher binv.
__global__ void __launch_bounds__(32)
k_argmax(const float* __restrict__ sfea, const float* __restrict__ tfea,
         const float* __restrict__ tfinv, int* __restrict__ ind) {
    int j = blockIdx.x;
    int t = j & 1;  j >>= 1;
    int b = j & 31; j >>= 5;
    int tg = j;                      // 0..49 tile slot
    int crop, tile, S;
    if (tg < 26) { crop = tg / 13; tile = tg % 13; S = NG; }
    else         { crop = 2 + (tg - 26) / 3; tile = (tg - 26) % 3; S = NL; }
    if (crop == t) return;           // whole wave exits; WMMA below runs with full EXEC
    int offs = (crop < 2) ? crop * BATCH * NG : 2 * BATCH * NG + (crop - 2) * BATCH * NL;
    int rowBase = offs + b * S;

    int lane = threadIdx.x;
    int m = lane & 15;
    int koff = (lane < 16) ? 0 : 2;  // A 16x4 f32 layout: lanes 16-31 carry K=2,3
    int aRow = tile * 16 + m;
    int aClamp = (aRow < S) ? aRow : (S - 1);
    const float* Ap = sfea + (size_t)(rowBase + aClamp) * FEA + koff;
    int tBase = (t * BATCH + b) * NG;

    float best[8]; int bidx[8];
#pragma unroll
    for (int i = 0; i < 8; ++i) { best[i] = -INFINITY; bidx[i] = 0; }

    for (int nt = 0; nt < 13; ++nt) {              // 13 column tiles cover 196 (+pad)
        int n  = nt * 16 + m;
        int nC = (n < NG) ? n : (NG - 1);
        const float* Bp = tfea + (size_t)(tBase + nC) * FEA + koff;
        float binv = tfinv[tBase + nC];
        v8f c = {0.f, 0.f, 0.f, 0.f, 0.f, 0.f, 0.f, 0.f};
#pragma unroll 4
        for (int k = 0; k < FEA; k += 4) {
            v2f a  = *(const v2f*)(Ap + k);
            v2f bb = *(const v2f*)(Bp + k);
            c = __builtin_amdgcn_wmma_f32_16x16x4_f32(false, a, false, bb,
                                                      (short)0, c, false, false);
        }
        bool valid = (n < NG);
#pragma unroll
        for (int i = 0; i < 8; ++i) {
            float v = valid ? (c[i] * binv) : -INFINITY;
            if (v > best[i]) { best[i] = v; bidx[i] = n; }
        }
    }
    // reduce over the 16 lanes of each column group (xor<=8 stays inside halves)
#pragma unroll
    for (int off = 1; off < 16; off <<= 1) {
#pragma unroll
        for (int i = 0; i < 8; ++i) {
            float ov = __shfl_xor(best[i], off, 32);
            int   oi = __shfl_xor(bidx[i], off, 32);
            if (ov > best[i] || (ov == best[i] && oi < bidx[i])) { best[i] = ov; bidx[i] = oi; }
        }
    }
    if (m == 0) {   // lane 0 -> rows 0..7, lane 16 -> rows 8..15 (C layout)
        int rbase = tile * 16 + ((lane >> 4) << 3);
#pragma unroll
        for (int i = 0; i < 8; ++i) {
            int row = rbase + i;
            if (row < S) ind[t * SROWS + rowBase + row] = bidx[i];
        }
    }
}

// ---------------- cls CE + L1 per (t, s, b) ------------------------------------
__global__ void k_cls(const float* __restrict__ scls, const float* __restrict__ tcls,
                      const float* __restrict__ center, const int* __restrict__ epoch,
                      const float* __restrict__ tm, const float* __restrict__ tinvZ,
                      const float* __restrict__ slse, double* __restrict__ acc) {
    int j = blockIdx.x;
    int b = j & 31; j >>= 5;
    int s = j % 10; int t = j / 10;
    if (s == t) return;
    float invT = 1.0f / temp_from_epoch(epoch[0]);
    int rT = t * BATCH + b;
    int rS = s * BATCH + b;
    const float* xt = tcls + (size_t)rT * K_DIM;
    const float* xs = scls + (size_t)rS * K_DIM;
    float mT = tm[rT], izT = tinvZ[rT];
    float lse = slse[rS];
    __shared__ float sm[256];
    int tid = threadIdx.x;
    float ce = 0.f, l1 = 0.f;
    for (int k = tid; k < K_DIM; k += 256) {
        float T  = __expf((xt[k] - center[k]) * invT - mT) * izT;
        float Sv = xs[k] * 10.0f;
        ce += T * (Sv - lse);
        l1 += fabsf(T - Sv);
    }
    sm[tid] = ce; __syncthreads();
    for (int r = 128; r > 0; r >>= 1) { if (tid < r) sm[tid] += sm[tid + r]; __syncthreads(); }
    ce = sm[0]; __syncthreads();
    sm[tid] = l1; __syncthreads();
    for (int r = 128; r > 0; r >>= 1) { if (tid < r) sm[tid] += sm[tid + r]; __syncthreads(); }
    l1 = sm[0];
    if (tid == 0) {
        double contrib = (double)(-0.5 / (BATCH * 18.0)) * (double)ce
                       + (double)(0.005 / ((double)BATCH * K_DIM * 18.0)) * (double)l1;
        atomicAdd(acc, contrib);
    }
}

// ---------------- region CE with gathered teacher row --------------------------
__global__ void k_region(const float* __restrict__ sreg, const float* __restrict__ treg,
                         const float* __restrict__ cgrid, const int* __restrict__ epoch,
                         const float* __restrict__ tm, const float* __restrict__ tinvZ,
                         const float* __restrict__ slse, const int* __restrict__ ind,
                         double* __restrict__ acc) {
    int bid = blockIdx.x;
    int t = bid & 1;
    int row = bid >> 1;
    int crop, b, S;
    decode_srow(row, crop, b, S);
    if (crop == t) return;
    float invT = 1.0f / temp_from_epoch(epoch[0]);
    int n  = ind[t * SROWS + row];
    int rT = (t * BATCH + b) * NG + n;
    const float* xt = treg + (size_t)rT * K_DIM;
    const float* xs = sreg + (size_t)row * K_DIM;
    float mT = tm[TCLS + rT], izT = tinvZ[TCLS + rT];
    float lse = slse[SCLS + row];
    __shared__ float sm[256];
    int tid = threadIdx.x;
    float ce = 0.f;
    for (int k = tid; k < K_DIM; k += 256) {
        float T = __expf((xt[k] - cgrid[k]) * invT - mT) * izT;
        ce += T * (xs[k] * 10.0f - lse);
    }
    sm[tid] = ce; __syncthreads();
    for (int r = 128; r > 0; r >>= 1) { if (tid < r) sm[tid] += sm[tid + r]; __syncthreads(); }
    if (tid == 0) {
        double w = -0.5 / ((double)S * BATCH * 18.0);
        atomicAdd(acc, w * (double)sm[0]);
    }
}

__global__ void k_init(double* acc) { if (threadIdx.x == 0 && blockIdx.x == 0) *acc = 0.0; }
__global__ void k_final(const double* __restrict__ acc, float* __restrict__ out) {
    if (threadIdx.x == 0 && blockIdx.x == 0) out[0] = (float)(*acc);
}

extern "C" void kernel_launch(void* const* d_in, const int* in_sizes, int n_in,
                              void* d_out, int out_size, void* d_ws, size_t ws_size,
                              hipStream_t stream) {
    const float* scls   = (const float*)d_in[0];  // [320, 4096]
    const float* sreg   = (const float*)d_in[1];  // [21760, 4096]
    const float* sfea   = (const float*)d_in[2];  // [21760, 384]
    const float* tcls   = (const float*)d_in[3];  // [64, 4096]
    const float* treg   = (const float*)d_in[4];  // [12544, 4096]
    const float* tfea   = (const float*)d_in[5];  // [12544, 384]
    const float* center = (const float*)d_in[6];  // [1, 4096]
    const float* cgrid  = (const float*)d_in[7];  // [1, 4096]
    const int*   epoch  = (const int*)d_in[8];    // scalar

    // workspace layout (~414 KB): acc(double) | tm | tinvZ | slse | tfinv | ind
    double* acc  = (double*)d_ws;
    float* fp    = (float*)((char*)d_ws + 16);
    float* tm    = fp;                         // 12608
    float* tinvZ = tm + (TCLS + TROWS);        // 12608
    float* slse  = tinvZ + (TCLS + TROWS);     // 22080
    float* tfinv = slse + (SCLS + SROWS);      // 12544
    int*   ind   = (int*)(tfinv + TROWS);      // 2*21760 ints

    k_init<<<1, 32, 0, stream>>>(acc);
    k_teacher_stats<<<TCLS + TROWS, 256, 0, stream>>>(tcls, treg, center, cgrid, epoch, tm, tinvZ);
    k_student_lse<<<SCLS + SROWS, 256, 0, stream>>>(scls, sreg, slse);
    k_tfea_norm<<<TROWS, 128, 0, stream>>>(tfea, tfinv);
    k_argmax<<<50 * BATCH * 2, 32, 0, stream>>>(sfea, tfea, tfinv, ind);
    k_cls<<<2 * 10 * BATCH, 256, 0, stream>>>(scls, tcls, center, epoch, tm, tinvZ, slse, acc);
    k_region<<<2 * SROWS, 256, 0, stream>>>(sreg, treg, cgrid, epoch, tm, tinvZ, slse, ind, acc);
    k_final<<<1, 32, 0, stream>>>(acc, (float*)d_out);
}